// DGN_54932631715985
// MI455X (gfx1250) — compile-verified
//
#include <hip/hip_runtime.h>

typedef __attribute__((ext_vector_type(2))) float v2f;
typedef __attribute__((ext_vector_type(8))) float v8f;

#define T_DIM 8
#define N_DIM 50000
#define E_DIM 800000
#define D_DIM 64

// ---------------------------------------------------------------------------
// Kernel 1: zero the output buffer (it is reused as the agg accumulator).
// ---------------------------------------------------------------------------
__global__ void dgn_zero_kernel(float4* __restrict__ out, int n4) {
    int i = blockIdx.x * blockDim.x + threadIdx.x;
    if (i < n4) out[i] = make_float4(0.f, 0.f, 0.f, 0.f);
}

// ---------------------------------------------------------------------------
// Kernel 2: temporal edge scatter.  One wave32 per (t, edge).
// Per-edge scalars are wave-uniform -> SMEM loads; the 64-float row of x is
// read with 2 coalesced lane loads; accumulation uses hardware f32 atomics.
// x[t] slice (12.8 MB) is L2-resident (192 MB L2), so gathers are cheap.
// ---------------------------------------------------------------------------
__global__ void dgn_scatter_kernel(const float* __restrict__ x,     // [T,N,64]
                                   const int*   __restrict__ ei,    // [2,E]
                                   const float* __restrict__ etime, // [E]
                                   const float* __restrict__ ntime, // [T]
                                   const float* __restrict__ ew,    // [E]
                                   float* out,                      // [T,N,64]
                                   int total_waves) {
    int wave = blockIdx.x * (blockDim.x >> 5) + (threadIdx.x >> 5);
    if (wave >= total_waves) return;
    int lane = threadIdx.x & 31;

    int t = wave / E_DIM;           // wave-uniform
    int e = wave - t * E_DIM;       // wave-uniform

    float te = etime[e];            // scalar loads (uniform address)
    float tn = ntime[t];
    if (te > tn) return;            // whole-wave uniform predicate

    int   src = ei[e];
    int   dst = ei[E_DIM + e];
    float w   = ew[e];

    size_t xb = ((size_t)t * N_DIM + (size_t)src) * D_DIM;
    size_t ob = ((size_t)t * N_DIM + (size_t)dst) * D_DIM;

    float v0 = x[xb + lane]      * w;
    float v1 = x[xb + lane + 32] * w;

    // hardware global_atomic_add_f32 (no CAS loop)
    unsafeAtomicAdd(&out[ob + lane],      v0);
    unsafeAtomicAdd(&out[ob + lane + 32], v1);
}

// ---------------------------------------------------------------------------
// Kernel 3: in-place  out[t,r,:] = agg[t,r,:] @ W + b  via V_WMMA_F32_16X16X4_F32.
// One wave per 16-row group; W is staged in LDS *pre-paired* along K so each
// B fragment is one aligned ds_load_b64 straight into the operand VGPR pair
// (eliminates the ds_load_2addr + v_dual_mov repacking seen previously).
//
// Fragment layouts (CDNA5 ISA 7.12.2):
//   A 16x4 : lane&15 = M ; VGPRs {K, K+1} with K = (lane>>4)*2
//   B 4x16 : lane&15 = N ; same K split
//   C/D    : VGPR j -> row j (lanes 0-15) / row j+8 (lanes 16-31), col = lane&15
// ---------------------------------------------------------------------------
__global__ void dgn_gemm_bias_kernel(float* out,                       // [T,N,64] in:agg out:result
                                     const float* __restrict__ W,      // [64,64]
                                     const float* __restrict__ bias,   // [64]
                                     int total_waves) {
    // sWp[k2*64 + n] = { W[2*k2][n], W[2*k2+1][n] },  k2 in [0,32)
    __shared__ v2f   sWp[(D_DIM / 2) * D_DIM];
    __shared__ float sB[D_DIM];

    for (int i = threadIdx.x; i < (D_DIM / 2) * D_DIM; i += blockDim.x) {
        int k2 = i >> 6;       // i / 64
        int n  = i & 63;       // i % 64
        v2f p;
        p.x = W[(2 * k2)     * D_DIM + n];
        p.y = W[(2 * k2 + 1) * D_DIM + n];
        sWp[i] = p;
    }
    if (threadIdx.x < D_DIM) sB[threadIdx.x] = bias[threadIdx.x];
    __syncthreads();

    int wave = blockIdx.x * (blockDim.x >> 5) + (threadIdx.x >> 5);
    if (wave >= total_waves) return;
    int lane = threadIdx.x & 31;

    const int groups_per_t = N_DIM / 16;      // 3125
    int t = wave / groups_per_t;
    int g = wave - t * groups_per_t;
    size_t base = ((size_t)t * N_DIM + (size_t)g * 16) * D_DIM;

    int m  = lane & 15;          // A: row M / B: col N / C: col N
    int hi = lane >> 4;          // 0 or 1 (selects K sub-pair)
    int kk = hi << 1;            // K sub-offset: 0 or 2

    v8f acc[4] = {};

    for (int k0 = 0; k0 < D_DIM; k0 += 4) {
        v2f a;
        const float* arow = out + base + (size_t)m * D_DIM + (k0 + kk);
        a.x = arow[0];           // contiguous pair -> global_load_b64
        a.y = arow[1];
        int k2row = ((k0 >> 1) + hi) * D_DIM;   // pre-paired LDS row
#pragma unroll
        for (int nt = 0; nt < 4; ++nt) {
            v2f bf = sWp[k2row + nt * 16 + m];  // single ds_load_b64
            acc[nt] = __builtin_amdgcn_wmma_f32_16x16x4_f32(
                /*neg_a=*/false, a, /*neg_b=*/false, bf,
                /*c_mod=*/(short)0, acc[nt],
                /*reuse_a=*/false, /*reuse_b=*/false);
        }
    }

    // bias add + in-place store (all reads above complete before these writes
    // in wave program order; row groups never overlap across waves).
    int half = hi << 3;   // 0 or 8
#pragma unroll
    for (int nt = 0; nt < 4; ++nt) {
        float bb = sB[nt * 16 + m];
#pragma unroll
        for (int i = 0; i < 8; ++i) {
            out[base + (size_t)(i + half) * D_DIM + nt * 16 + m] = acc[nt][i] + bb;
        }
    }
}

// ---------------------------------------------------------------------------
extern "C" void kernel_launch(void* const* d_in, const int* in_sizes, int n_in,
                              void* d_out, int out_size, void* d_ws, size_t ws_size,
                              hipStream_t stream) {
    const float* x     = (const float*)d_in[0];  // [T,N,64]
    const int*   ei    = (const int*)  d_in[1];  // [2,E]
    const float* etime = (const float*)d_in[2];  // [E]
    const float* ntime = (const float*)d_in[3];  // [T]
    const float* ew    = (const float*)d_in[4];  // [E]
    const float* W     = (const float*)d_in[5];  // [64,64]
    const float* b     = (const float*)d_in[6];  // [64]
    float* out = (float*)d_out;                  // [T,N,64]

    // 1) zero accumulator (d_out)
    int n4 = (T_DIM * N_DIM * D_DIM) / 4;        // 6.4M float4
    dgn_zero_kernel<<<(n4 + 255) / 256, 256, 0, stream>>>((float4*)out, n4);

    // 2) temporal edge scatter: one wave per (t, edge)
    int scat_waves  = T_DIM * E_DIM;             // 6.4M waves
    int scat_blocks = (scat_waves + 7) / 8;      // 8 waves / 256-thread block
    dgn_scatter_kernel<<<scat_blocks, 256, 0, stream>>>(x, ei, etime, ntime, ew,
                                                        out, scat_waves);

    // 3) in-place WMMA GEMM + bias: one wave per 16-row group
    int gemm_waves  = T_DIM * (N_DIM / 16);      // 25000
    int gemm_blocks = (gemm_waves + 7) / 8;
    dgn_gemm_bias_kernel<<<gemm_blocks, 256, 0, stream>>>(out, W, b, gemm_waves);
}